// BlockAttention_58110907515325
// MI455X (gfx1250) — compile-verified
//
#include <hip/hip_runtime.h>
#include <hip/hip_bf16.h>

// Shapes fixed by the reference: B=32, C=256, H=W=128, TOP_K=16.
#define B 32
#define C 256
#define HW (128 * 128)        // 16384 floats per channel plane
#define TOPK 16

typedef __attribute__((ext_vector_type(2))) float v2f;
typedef __attribute__((ext_vector_type(8))) float v8f;

// ---------------------------------------------------------------------------
// Kernel 1: global average pool.  One 256-thread block per (b,c) plane.
// Reads 512 MB once (HBM-bound: ~22 us at 23.3 TB/s).
// ---------------------------------------------------------------------------
__global__ __launch_bounds__(256) void pool_kernel(const float* __restrict__ x,
                                                   float* __restrict__ y) {
  const int plane = blockIdx.x;  // b*C + c
  const float4* p = (const float4*)(x + (size_t)plane * HW);
  float s = 0.0f;
  // 4096 float4 per plane / 256 threads = 16 iterations
  for (int i = threadIdx.x; i < HW / 4; i += 256) {
    float4 v = p[i];
    s += (v.x + v.y) + (v.z + v.w);
  }
  __shared__ float red[256];
  red[threadIdx.x] = s;
  __syncthreads();
  for (int off = 128; off > 0; off >>= 1) {
    if (threadIdx.x < off) red[threadIdx.x] += red[threadIdx.x + off];
    __syncthreads();
  }
  if (threadIdx.x == 0) y[plane] = red[0] * (1.0f / (float)HW);
}

// ---------------------------------------------------------------------------
// Kernel 2: gate MLP with fp32 WMMA (V_WMMA_F32_16X16X4_F32) + per-sample
// top-16.  Single workgroup, 512 threads = 16 wave32.
//
// GEMM: out[m,n] = sum_k In[m,k] * W[n,k]   (In @ W^T), M=32, N=256, K=256.
// Tiles: 2 M-tiles x 16 N-tiles = 32 tasks over 16 waves (2 tasks each),
// K consumed 4 at a time by the WMMA.
//
// fp32 A-fragment layout (ISA 7.12.2, 16x4 f32): lane<16 -> K=k0+{0,1},
// lane>=16 -> K=k0+{2,3}, row M = lane%16.  B (4x16) mirrors with N = lane%16.
// C/D (16x16 f32): VGPR r holds M = r + 8*(lane>=16), N = lane%16.
// ---------------------------------------------------------------------------
__global__ __launch_bounds__(512) void gate_topk_kernel(
    const float* __restrict__ Y,    // [B,C] pooled means (workspace)
    const float* __restrict__ W1,   // [C,C]
    const float* __restrict__ W2,   // [C,C]
    int* __restrict__ idx_out) {    // [B,TOPK]
  __shared__ float sA[B * C];  // holds Y, later reused for gates G
  __shared__ float sB[B * C];  // holds hidden H

  // Stage Y into LDS: 8192 floats / 512 threads = 16 each.
  for (int i = threadIdx.x; i < B * C; i += 512) sA[i] = Y[i];
  __syncthreads();

  const int lane  = threadIdx.x & 31;
  const int wave  = threadIdx.x >> 5;
  const int mrow  = lane & 15;        // M (for A) / N (for B,C,D) sub-index
  const int khalf = lane >> 4;        // 0: K+{0,1}, 1: K+{2,3}

  // ---------------- Layer 1: H = relu(Y @ W1^T) ----------------
  for (int task = wave; task < 32; task += 16) {
    const int mt = task >> 4;         // 0..1
    const int nt = task & 15;         // 0..15
    const float* in_row = sA + (mt * 16 + mrow) * C;        // A row (M = mrow)
    const float* w_row  = W1 + (size_t)(nt * 16 + mrow) * C; // B col n = mrow
    v8f acc = {};
    for (int k0 = 0; k0 < C; k0 += 4) {
      const int kk = k0 + 2 * khalf;
      v2f a = *(const v2f*)(in_row + kk);
      v2f b = *(const v2f*)(w_row + kk);
      acc = __builtin_amdgcn_wmma_f32_16x16x4_f32(
          false, a, false, b, (short)0, acc, false, false);
    }
    #pragma unroll
    for (int r = 0; r < 8; ++r) {
      float v = acc[r];
      v = v > 0.0f ? v : 0.0f;  // ReLU
      const int mm = mt * 16 + r + 8 * khalf;
      sB[mm * C + nt * 16 + mrow] = v;
    }
  }
  __syncthreads();

  // ---------------- Layer 2: G = sigmoid(H @ W2^T) ----------------
  for (int task = wave; task < 32; task += 16) {
    const int mt = task >> 4;
    const int nt = task & 15;
    const float* in_row = sB + (mt * 16 + mrow) * C;
    const float* w_row  = W2 + (size_t)(nt * 16 + mrow) * C;
    v8f acc = {};
    for (int k0 = 0; k0 < C; k0 += 4) {
      const int kk = k0 + 2 * khalf;
      v2f a = *(const v2f*)(in_row + kk);
      v2f b = *(const v2f*)(w_row + kk);
      acc = __builtin_amdgcn_wmma_f32_16x16x4_f32(
          false, a, false, b, (short)0, acc, false, false);
    }
    #pragma unroll
    for (int r = 0; r < 8; ++r) {
      const float v = acc[r];
      const float g = 1.0f / (1.0f + __expf(-v));  // sigmoid (monotone)
      const int mm = mt * 16 + r + 8 * khalf;
      sA[mm * C + nt * 16 + mrow] = g;
    }
  }
  __syncthreads();

  // ---------------- Top-16 per sample (matches jax.lax.top_k ordering:
  // descending value, ties broken by lowest index via strict '>') ----------
  if (threadIdx.x < B) {
    const int bidx = threadIdx.x;
    float* g = sA + bidx * C;
    for (int k = 0; k < TOPK; ++k) {
      float best = -3.0e38f;
      int bi = 0;
      for (int c = 0; c < C; ++c) {
        const float v = g[c];
        if (v > best) { best = v; bi = c; }
      }
      idx_out[bidx * TOPK + k] = bi;
      g[bi] = -3.0e38f;
    }
  }
}

// ---------------------------------------------------------------------------
// Kernel 3: gather selected channels.  One 256-thread block per (b,k) plane.
// ---------------------------------------------------------------------------
__global__ __launch_bounds__(256) void gather_kernel(
    const float* __restrict__ x, const int* __restrict__ idx,
    float* __restrict__ out) {
  const int plane = blockIdx.x;       // b*TOPK + k
  const int b = plane >> 4;
  const int c = idx[plane];
  const float4* src = (const float4*)(x + ((size_t)b * C + c) * HW);
  float4* dst = (float4*)(out + (size_t)plane * HW);
  for (int i = threadIdx.x; i < HW / 4; i += 256) dst[i] = src[i];
}

// ---------------------------------------------------------------------------
extern "C" void kernel_launch(void* const* d_in, const int* in_sizes, int n_in,
                              void* d_out, int out_size, void* d_ws,
                              size_t ws_size, hipStream_t stream) {
  const float* x  = (const float*)d_in[0];
  const float* W1 = (const float*)d_in[1];
  const float* W2 = (const float*)d_in[2];
  float* out = (float*)d_out;

  float* y  = (float*)d_ws;                               // B*C floats
  int* idx  = (int*)((char*)d_ws + B * C * sizeof(float)); // B*TOPK ints

  pool_kernel<<<B * C, 256, 0, stream>>>(x, y);
  gate_topk_kernel<<<1, 512, 0, stream>>>(y, W1, W2, idx);
  gather_kernel<<<B * TOPK, 256, 0, stream>>>(x, idx, out);
}